// CAGL_69784628626150
// MI455X (gfx1250) — compile-verified
//
#include <hip/hip_runtime.h>
#include <stdint.h>

#define B_   64
#define R_   8
#define W_   30000
#define DV_  1024
#define TDV_ 2048
#define K_   32
#define L_   20
#define C_   1000
#define BETA_REL 0.3f
#define NCHUNK 15
#define CHUNK  2000   // 15*2000 == 30000, divisible by 4

typedef __attribute__((ext_vector_type(2))) float v2f;
typedef __attribute__((ext_vector_type(8))) float v8f;

// ---------------- Kernel A: per-(b,r) min/max over W ----------------
__global__ void k_minmax(const float* __restrict__ x,
                         float* __restrict__ mn, float* __restrict__ mx) {
    int row = blockIdx.x;                 // b*R + r
    const float4* p = (const float4*)(x + (size_t)row * W_);
    float lmn = __builtin_inff(), lmx = -__builtin_inff();
    for (int i = threadIdx.x; i < W_/4; i += blockDim.x) {
        float4 v = p[i];
        lmn = fminf(lmn, fminf(fminf(v.x, v.y), fminf(v.z, v.w)));
        lmx = fmaxf(lmx, fmaxf(fmaxf(v.x, v.y), fmaxf(v.z, v.w)));
    }
    __shared__ float smn[256], smx[256];
    smn[threadIdx.x] = lmn; smx[threadIdx.x] = lmx;
    __syncthreads();
    for (int s = blockDim.x >> 1; s > 0; s >>= 1) {
        if (threadIdx.x < s) {
            smn[threadIdx.x] = fminf(smn[threadIdx.x], smn[threadIdx.x + s]);
            smx[threadIdx.x] = fmaxf(smx[threadIdx.x], smx[threadIdx.x + s]);
        }
        __syncthreads();
    }
    if (threadIdx.x == 0) { mn[row] = smn[0]; mx[row] = smx[0]; }
}

// ------- Kernel B: e = exp(max_r (x-mn)/(mx-mn)), partial row sums -------
__global__ void k_modelpre(const float* __restrict__ x,
                           const float* __restrict__ mn, const float* __restrict__ mx,
                           float* __restrict__ ebuf, float* __restrict__ partial) {
    int chunk = blockIdx.x, b = blockIdx.y;
    int w0 = chunk * CHUNK;
    float lo[R_], inv[R_];
    for (int r = 0; r < R_; ++r) {
        float a = mn[b*R_ + r], c = mx[b*R_ + r];
        lo[r] = a; inv[r] = 1.0f / (c - a);
    }
    float lsum = 0.0f;
    for (int i4 = threadIdx.x; i4 < CHUNK/4; i4 += blockDim.x) {
        int w = w0 + i4*4;
        float4 best = make_float4(-__builtin_inff(), -__builtin_inff(),
                                  -__builtin_inff(), -__builtin_inff());
        for (int r = 0; r < R_; ++r) {
            float4 v = *(const float4*)(x + ((size_t)(b*R_ + r))*W_ + w);
            best.x = fmaxf(best.x, (v.x - lo[r]) * inv[r]);
            best.y = fmaxf(best.y, (v.y - lo[r]) * inv[r]);
            best.z = fmaxf(best.z, (v.z - lo[r]) * inv[r]);
            best.w = fmaxf(best.w, (v.w - lo[r]) * inv[r]);
        }
        float4 e = make_float4(expf(best.x), expf(best.y), expf(best.z), expf(best.w));
        *(float4*)(ebuf + (size_t)b*W_ + w) = e;
        lsum += e.x + e.y + e.z + e.w;
    }
    __shared__ float ss[256];
    ss[threadIdx.x] = lsum;
    __syncthreads();
    for (int s = blockDim.x >> 1; s > 0; s >>= 1) {
        if (threadIdx.x < s) ss[threadIdx.x] += ss[threadIdx.x + s];
        __syncthreads();
    }
    if (threadIdx.x == 0) partial[b*NCHUNK + chunk] = ss[0];
}

// -- Kernel C: refine = 0.5*e/Zm + 0.5*exp(count)/Zc (in place over ebuf) --
__global__ void k_refine(const int* __restrict__ dw, float* __restrict__ ebuf,
                         const float* __restrict__ partial) {
    int b = blockIdx.x;
    __shared__ float sAm, sBase, sBonus[L_];
    __shared__ int   sFirst[L_];
    if (threadIdx.x == 0) {
        float zm = 0.0f;
        for (int i = 0; i < NCHUNK; ++i) zm += partial[b*NCHUNK + i];
        float zc = 0.0f; int ndist = 0;
        for (int t = 0; t < L_; ++t) {
            int wt = dw[b*L_ + t];
            int cnt = 0; bool first = true;
            for (int j = 0; j < L_; ++j)
                if (dw[b*L_ + j] == wt) { ++cnt; if (j < t) first = false; }
            sFirst[t] = first ? 1 : 0;
            float ex = expf((float)cnt);
            if (first) { zc += ex; ++ndist; sBonus[t] = ex; } else sBonus[t] = 0.0f;
        }
        zc += (float)(W_ - ndist);
        sAm = 0.5f / zm;
        sBase = 0.5f / zc;
        for (int t = 0; t < L_; ++t)
            sBonus[t] = sFirst[t] ? 0.5f * (sBonus[t] - 1.0f) / zc : 0.0f;
    }
    __syncthreads();
    float am = sAm, base = sBase;
    float* eb = ebuf + (size_t)b * W_;
    for (int i = threadIdx.x; i < W_; i += blockDim.x)
        eb[i] = am * eb[i] + base;
    __syncthreads();
    if (threadIdx.x < L_ && sFirst[threadIdx.x])
        eb[dw[b*L_ + threadIdx.x]] += sBonus[threadIdx.x];
}

// ---- Kernel D: exact top-32 by iterative argmax (lax.top_k tie order) ----
__global__ void k_topk(float* __restrict__ ebuf, int* __restrict__ topk) {
    int b = blockIdx.x;
    float* eb = ebuf + (size_t)b * W_;
    __shared__ unsigned long long skey[1024];
    for (int it = 0; it < K_; ++it) {
        unsigned long long best = 0ull;
        for (int i = threadIdx.x; i < W_; i += blockDim.x) {
            float v = eb[i];
            if (v >= 0.0f) {   // refine>0 always; masked entries are -1
                unsigned int fb = __float_as_uint(v);
                unsigned long long key =
                    ((unsigned long long)fb << 32) | (unsigned int)(0xFFFFFFFFu - (unsigned)i);
                if (key > best) best = key;
            }
        }
        skey[threadIdx.x] = best;
        __syncthreads();
        for (int s = blockDim.x >> 1; s > 0; s >>= 1) {
            if (threadIdx.x < s) {
                unsigned long long o = skey[threadIdx.x + s];
                if (o > skey[threadIdx.x]) skey[threadIdx.x] = o;
            }
            __syncthreads();
        }
        if (threadIdx.x == 0) {
            int idx = (int)(0xFFFFFFFFu - (unsigned int)(skey[0] & 0xFFFFFFFFu));
            topk[b*K_ + it] = idx;
            eb[idx] = -1.0f;
        }
        __syncthreads();
    }
}

// ---- Kernel E: closed-form w_eff = fuse_w^T adj_norm; mix_fuse directly ----
__global__ void k_fuse(const int* __restrict__ topk, const int* __restrict__ dw,
                       const float* __restrict__ embed, const float* __restrict__ featv,
                       const float* __restrict__ fw, const float* __restrict__ fbp,
                       float* __restrict__ mix_fuse) {
    int b = blockIdx.x;
    __shared__ int   sid[K_];
    __shared__ float sm[K_], sweff[K_], ssw;
    int t = threadIdx.x;
    if (t < K_) {
        int id = topk[b*K_ + t];
        sid[t] = id;
        float mm = 0.0f;
        for (int j = 0; j < L_; ++j) if (dw[b*L_ + j] == id) mm = 1.0f;
        sm[t] = mm;
    }
    __syncthreads();
    if (t < K_) {
        float Sm = 0.0f;
        for (int i = 0; i < K_; ++i) Sm += sm[i];
        float S1 = 0.0f, S2 = 0.0f;
        for (int i = 0; i < K_; ++i) {
            float deg = 1.0f + BETA_REL*(K_-1) + BETA_REL*sm[i]*Sm;
            float f = fw[i] / deg;
            S1 += f; S2 += f * sm[i];
        }
        float degj = 1.0f + BETA_REL*(K_-1) + BETA_REL*sm[t]*Sm;
        sweff[t] = BETA_REL*S1 + (1.0f-BETA_REL)*fw[t]/degj + BETA_REL*sm[t]*S2;
    }
    __syncthreads();
    if (t == 0) {
        float s = 0.0f;
        for (int i = 0; i < K_; ++i) s += sweff[i];
        ssw = s;
    }
    __syncthreads();
    float fb = fbp[0];
    for (int d = t; d < DV_; d += blockDim.x) {
        float acc = 0.0f;
        for (int j = 0; j < K_; ++j)
            acc += sweff[j] * embed[(size_t)sid[j]*DV_ + d];
        mix_fuse[(size_t)b*TDV_ + d] = acc + fb;
    }
    float sw = ssw;
    for (int d = t; d < DV_; d += blockDim.x)
        mix_fuse[(size_t)b*TDV_ + DV_ + d] = sw * featv[(size_t)b*DV_ + d] + fb;
}

// ---- Kernel F: classifier GEMM [64,2048]x[2048,1000] via WMMA f32 16x16x4 ----
__global__ void __launch_bounds__(128)
k_cls(const float* __restrict__ mixf, const float* __restrict__ cw,
      const float* __restrict__ cb, float* __restrict__ out) {
    int wave  = threadIdx.x >> 5;     // 0..3 -> m tile
    int lane  = threadIdx.x & 31;
    int khalf = lane >> 4;            // 0: K={0,1}, 1: K={2,3}
    int lo16  = lane & 15;
    int n0    = blockIdx.x * 16;

    int mrow = wave*16 + lo16;
    int ncol = n0 + lo16;
    int nclamp = ncol < C_ ? ncol : (C_ - 1);

    const float* arow = mixf + (size_t)mrow  * TDV_ + 2*khalf;   // A[m][k..k+1]
    const float* brow = cw   + (size_t)nclamp * TDV_ + 2*khalf;  // B[k..k+1][n]

    v8f acc = {};
    #pragma unroll 4
    for (int k = 0; k < TDV_; k += 4) {
        float2 av = *(const float2*)(arow + k);
        float2 bv = *(const float2*)(brow + k);
        v2f a; a.x = av.x; a.y = av.y;
        v2f b; b.x = bv.x; b.y = bv.y;
        acc = __builtin_amdgcn_wmma_f32_16x16x4_f32(
                  false, a, false, b, (short)0, acc, false, false);
    }
    if (ncol < C_) {
        float bias = cb[ncol];
        #pragma unroll
        for (int v = 0; v < 8; ++v) {
            int m = wave*16 + v + 8*khalf;    // C/D layout: VGPR v -> M=v / v+8
            out[(size_t)m * C_ + ncol] = acc[v] + bias;
        }
    }
}

extern "C" void kernel_launch(void* const* d_in, const int* in_sizes, int n_in,
                              void* d_out, int out_size, void* d_ws, size_t ws_size,
                              hipStream_t stream) {
    const float* predicts = (const float*)d_in[0];
    const float* featv    = (const float*)d_in[1];
    const int*   dw       = (const int*)d_in[2];
    const float* embed    = (const float*)d_in[3];
    const float* fuse_w   = (const float*)d_in[4];
    const float* fuse_b   = (const float*)d_in[5];
    const float* cls_w    = (const float*)d_in[6];
    const float* cls_b    = (const float*)d_in[7];

    float* out        = (float*)d_out;
    float* mix_fuse   = out;                       // [64,2048]
    float* predicts_v = out + (size_t)B_ * TDV_;   // [64,1000]

    float* ws      = (float*)d_ws;
    float* ebuf    = ws;                              // B*W floats (7.68 MB)
    float* mn      = ebuf + (size_t)B_ * W_;          // 512
    float* mx      = mn + B_ * R_;                    // 512
    float* partial = mx + B_ * R_;                    // 960
    int*   topk    = (int*)(partial + B_ * NCHUNK);   // 2048 ints

    k_minmax  <<<B_ * R_,            256,  0, stream>>>(predicts, mn, mx);
    k_modelpre<<<dim3(NCHUNK, B_),   256,  0, stream>>>(predicts, mn, mx, ebuf, partial);
    k_refine  <<<B_,                 256,  0, stream>>>(dw, ebuf, partial);
    k_topk    <<<B_,                 1024, 0, stream>>>(ebuf, topk);
    k_fuse    <<<B_,                 256,  0, stream>>>(topk, dw, embed, featv,
                                                        fuse_w, fuse_b, mix_fuse);
    k_cls     <<<(C_ + 15) / 16,     128,  0, stream>>>(mix_fuse, cls_w, cls_b, predicts_v);
}